// NetworkGL_Lite_70033736728882
// MI455X (gfx1250) — compile-verified
//
#include <hip/hip_runtime.h>
#include <math.h>

#define Bk 64
#define Lk 512
#define Ck 321
#define BCk (Bk * Ck)          // 20544
#define Nk 64
#define PATCHk 16
#define HIDk 32
#define PREDk 96
#define CHUNKS 11              // ceil(321/32)

typedef float v2f __attribute__((ext_vector_type(2)));
typedef float v4f __attribute__((ext_vector_type(4)));
typedef float v8f __attribute__((ext_vector_type(8)));

__device__ __forceinline__ float gelu_exact(float v) {
    return 0.5f * v * (1.0f + erff(v * 0.70710678118654752440f));
}

// Two chained fp32 WMMAs consuming one 16B A-frag and one 16B B-frag.
// Slot map: lane-half hi holds physical k = k0 + 4*hi + {0,1} (lo) and
// + {2,3} (hi-pass); A and B use identical maps, so the k-permutation is
// consistent and the pair covers k0..k0+7 exactly once.
__device__ __forceinline__ v8f wmma_pair(v8f acc, v4f a, v4f b) {
    v2f alo = {a.x, a.y}, ahi = {a.z, a.w};
    v2f blo = {b.x, b.y}, bhi = {b.z, b.w};
    acc = __builtin_amdgcn_wmma_f32_16x16x4_f32(false, alo, false, blo,
                                                (short)0, acc, false, false);
    acc = __builtin_amdgcn_wmma_f32_16x16x4_f32(false, ahi, false, bhi,
                                                (short)0, acc, false, false);
    return acc;
}

// ---------------------------------------------------------------------------
// K1: RevIN stats + normalize + EMA trend + seasonal, transposed write.
// One wave handles (b, 32 channels); lane = channel -> coalesced x loads,
// EMA recurrence is serial per-lane. 32x32 LDS transpose for coalesced row
// writes of s(BC,512) and trend(BC,512).
// ---------------------------------------------------------------------------
__global__ void k_revin_ema(const float* __restrict__ x,
                            const float* __restrict__ rev_w,
                            const float* __restrict__ rev_b,
                            float* __restrict__ s, float* __restrict__ trend,
                            float* __restrict__ meanv, float* __restrict__ stdv) {
    __shared__ float sT[4][32][33];
    __shared__ float tT[4][32][33];
    int wave = threadIdx.x >> 5, lane = threadIdx.x & 31;
    int wt = blockIdx.x * 4 + wave;          // 0..703
    int b  = wt / CHUNKS;
    int c0 = (wt % CHUNKS) * 32;
    int c  = c0 + lane;
    bool valid = (c < Ck);
    const float* xp = x + (size_t)b * Lk * Ck + (valid ? c : 0);

    float sum = 0.f, sq = 0.f;
    for (int l = 0; l < Lk; ++l) {
        float v = xp[(size_t)l * Ck];
        sum += v; sq += v * v;
    }
    float mean = sum * (1.0f / Lk);
    float var  = (sq - (float)Lk * mean * mean) * (1.0f / (Lk - 1));
    var = var > 0.f ? var : 0.f;
    float sd = sqrtf(var) + 1e-5f;
    int row = b * Ck + c;
    if (valid) { meanv[row] = mean; stdv[row] = sd; }
    float inv = 1.0f / sd;
    float rw = valid ? rev_w[c] : 1.f;
    float rb = valid ? rev_b[c] : 0.f;

    float carry = 0.f;
    for (int cb = 0; cb < 16; ++cb) {
        for (int ll = 0; ll < 32; ++ll) {
            int l = cb * 32 + ll;
            float v  = xp[(size_t)l * Ck];
            float xn = (v - mean) * inv * rw + rb;
            float tr = (l == 0) ? xn : (0.2f * xn + 0.8f * carry);
            carry = tr;
            sT[wave][ll][lane] = xn - tr;
            tT[wave][ll][lane] = tr;
        }
        __syncthreads();
        for (int cc = 0; cc < 32; ++cc) {
            int ch = c0 + cc;
            if (ch < Ck) {
                size_t off = (size_t)(b * Ck + ch) * Lk + cb * 32 + lane;
                s[off]     = sT[wave][lane][cc];
                trend[off] = tT[wave][lane][cc];
            }
        }
        __syncthreads();
    }
}

// ---------------------------------------------------------------------------
// K2: h1 = gelu(sp @ fc1_w.T + fc1_b); per-row BN1 partial sums (deterministic)
// One block per row. thread t -> n = t>>2, j = (t&3)*8 + k.
// ---------------------------------------------------------------------------
__global__ void k_fc1(const float* __restrict__ s,
                      const float* __restrict__ fc1_w,  // (32,16)
                      const float* __restrict__ fc1_b,
                      float* __restrict__ h,            // (BC, 2048)
                      float* __restrict__ pSum, float* __restrict__ pSq) {
    __shared__ float srow[512];
    __shared__ float wsh[32][16];
    __shared__ float bsh[32];
    __shared__ float red1[256], red2[256];
    int t = threadIdx.x, row = blockIdx.x;
    for (int i = t; i < 512; i += 256) srow[i] = s[(size_t)row * 512 + i];
    for (int i = t; i < 512; i += 256) ((float*)wsh)[i] = fc1_w[i];
    if (t < 32) bsh[t] = fc1_b[t];
    __syncthreads();

    int n = t >> 2;
    int j0 = (t & 3) * 8;
    float spv[16];
#pragma unroll
    for (int p = 0; p < 16; ++p) {
        int idx = n * 8 + p; if (idx > 511) idx = 511;   // pad-repeat last value
        spv[p] = srow[idx];
    }
    float ps = 0.f, pq = 0.f;
#pragma unroll
    for (int k = 0; k < 8; ++k) {
        int j = j0 + k;
        float acc = bsh[j];
#pragma unroll
        for (int p = 0; p < 16; ++p) acc += spv[p] * wsh[j][p];
        float g = gelu_exact(acc);
        h[(size_t)row * 2048 + n * 32 + j] = g;
        ps += g; pq += g * g;
    }
    red1[t] = ps; red2[t] = pq;
    __syncthreads();
    if (t < 64) {
        int base = t * 4;
        float S = red1[base] + red1[base + 1] + red1[base + 2] + red1[base + 3];
        float Q = red2[base] + red2[base + 1] + red2[base + 2] + red2[base + 3];
        pSum[(size_t)t * BCk + row] = S;
        pSq [(size_t)t * BCk + row] = Q;
    }
}

// ---------------------------------------------------------------------------
// K3/K5: reduce per-row partials -> BN affine (scale, shift) per channel n.
// ---------------------------------------------------------------------------
__global__ void k_bnstats(const float* __restrict__ pSum, const float* __restrict__ pSq,
                          const float* __restrict__ bw, const float* __restrict__ bb,
                          float* __restrict__ bnS, float* __restrict__ bnO) {
    __shared__ float r1[256], r2[256];
    int n = blockIdx.x, t = threadIdx.x;
    float S = 0.f, Q = 0.f;
    for (int i = t; i < BCk; i += 256) {
        S += pSum[(size_t)n * BCk + i];
        Q += pSq [(size_t)n * BCk + i];
    }
    r1[t] = S; r2[t] = Q; __syncthreads();
    for (int off = 128; off > 0; off >>= 1) {
        if (t < off) { r1[t] += r1[t + off]; r2[t] += r2[t + off]; }
        __syncthreads();
    }
    if (t == 0) {
        float cnt = (float)BCk * 32.0f;
        float mu  = r1[0] / cnt;
        float var = r2[0] / cnt - mu * mu;            // biased, matches jnp.var
        float sc  = bw[n] * rsqrtf(var + 1e-5f);
        bnS[n] = sc; bnO[n] = bb[n] - mu * sc;
    }
}

// ---------------------------------------------------------------------------
// K4: apply BN1, depthwise conv3 (zero pad) + bias, GELU; BN2 partials.
// One thread per (row, n); in-place on h.
// ---------------------------------------------------------------------------
__global__ void k_conv(float* __restrict__ h,
                       const float* __restrict__ bn1S, const float* __restrict__ bn1O,
                       const float* __restrict__ conv_w,   // (64,1,3)
                       const float* __restrict__ conv_b,
                       float* __restrict__ pSum, float* __restrict__ pSq) {
    int gid = blockIdx.x * 256 + threadIdx.x;
    int row = gid >> 6, n = gid & 63;
    float* hp = h + (size_t)row * 2048 + n * 32;
    float v[32];
#pragma unroll
    for (int j = 0; j < 32; ++j) v[j] = hp[j];
    float sc = bn1S[n], of = bn1O[n];
#pragma unroll
    for (int j = 0; j < 32; ++j) v[j] = v[j] * sc + of;
    float w0 = conv_w[n * 3], w1 = conv_w[n * 3 + 1], w2 = conv_w[n * 3 + 2];
    float cb = conv_b[n];
    float ps = 0.f, pq = 0.f;
    float o[32];
#pragma unroll
    for (int j = 0; j < 32; ++j) {
        float a   = (j > 0)  ? v[j - 1] : 0.f;
        float cpo = (j < 31) ? v[j + 1] : 0.f;
        o[j] = a * w0 + v[j] * w1 + cpo * w2 + cb;
    }
#pragma unroll
    for (int j = 0; j < 32; ++j) {
        float g = gelu_exact(o[j]);
        hp[j] = g; ps += g; pq += g * g;
    }
    pSum[(size_t)n * BCk + row] = ps;
    pSq [(size_t)n * BCk + row] = pq;
}

// ---------------------------------------------------------------------------
// K6: BN2 apply + fc2 + gate MLP + combine + LayerNorm -> y (first 1024 of
// each 2048-float h row).  y = (g2bn@fc2^T + fc2_b) + sp*(3 + gs*w[n]).
// ---------------------------------------------------------------------------
__global__ void k_mix(const float* __restrict__ s,
                      float* __restrict__ h,
                      const float* __restrict__ bn2S, const float* __restrict__ bn2O,
                      const float* __restrict__ fc2_w,  // (16,32)
                      const float* __restrict__ fc2_b,
                      const float* __restrict__ m1_w, const float* __restrict__ m1_b,
                      const float* __restrict__ m2_w, const float* __restrict__ m2_b,
                      const float* __restrict__ gl_scale,
                      const float* __restrict__ ln_w, const float* __restrict__ ln_b) {
    __shared__ float g2[64][32];
    __shared__ float sp[64][16];
    __shared__ float pooled[64];
    __shared__ float u[128];
    __shared__ float wv[64];
    __shared__ float yt[64][16];
    __shared__ float lnS[64], lnO[64];
    __shared__ float w2sh[16][32];
    int t = threadIdx.x, row = blockIdx.x;
    for (int i = t; i < 2048; i += 256) {
        int n = i >> 5;
        ((float*)g2)[i] = h[(size_t)row * 2048 + i] * bn2S[n] + bn2O[n];
    }
    for (int i = t; i < 1024; i += 256) {
        int n = i >> 4, p = i & 15;
        int idx = n * 8 + p; if (idx > 511) idx = 511;
        ((float*)sp)[i] = s[(size_t)row * 512 + idx];
    }
    for (int i = t; i < 512; i += 256) ((float*)w2sh)[i] = fc2_w[i];
    __syncthreads();
    if (t < 64) {
        float a = 0.f;
#pragma unroll
        for (int p = 0; p < 16; ++p) a += sp[t][p];
        pooled[t] = a * (1.0f / 16.0f);
    }
    __syncthreads();
    if (t < 128) {
        float a = m1_b[t];
        for (int k = 0; k < 64; ++k) a += pooled[k] * m1_w[t * 64 + k];
        u[t] = gelu_exact(a);
    }
    __syncthreads();
    if (t < 64) {
        float a = m2_b[t];
        for (int k = 0; k < 128; ++k) a += u[k] * m2_w[t * 128 + k];
        wv[t] = 1.0f / (1.0f + expf(-a));
    }
    __syncthreads();
    float gs = gl_scale[0];
    int n = t >> 2, p0 = (t & 3) * 4;
#pragma unroll
    for (int pp = 0; pp < 4; ++pp) {
        int p = p0 + pp;
        float a = fc2_b[p];
#pragma unroll
        for (int k = 0; k < 32; ++k) a += g2[n][k] * w2sh[p][k];
        a += sp[n][p] * (3.0f + gs * wv[n]);   // sp + (fc2+sp) + sp*(1+gs*w)
        yt[n][p] = a;
    }
    __syncthreads();
    if (t < 64) {
        float mu = 0.f;
#pragma unroll
        for (int p = 0; p < 16; ++p) mu += yt[t][p];
        mu *= (1.0f / 16.0f);
        float var = 0.f;
#pragma unroll
        for (int p = 0; p < 16; ++p) { float d = yt[t][p] - mu; var += d * d; }
        var *= (1.0f / 16.0f);
        lnS[t] = rsqrtf(var + 1e-5f); lnO[t] = mu;
    }
    __syncthreads();
#pragma unroll
    for (int pp = 0; pp < 4; ++pp) {
        int p = p0 + pp;
        float yv = (yt[n][p] - lnO[n]) * lnS[n] * ln_w[p] + ln_b[p];
        h[(size_t)row * 2048 + n * 16 + p] = yv;
    }
}

// ---------------------------------------------------------------------------
// fp32 WMMA GEMM: Z[rows, zoff:zoff+96] = A(rows,K) @ W(96,K)^T + bias.
// Wave handles a 16x96 tile. 16B-per-lane loads cover 8 k's; double-buffered
// so next iteration's 7 loads issue before the current 12 WMMAs.
// ---------------------------------------------------------------------------
__global__ void k_gemm96(const float* __restrict__ A, int lda,
                         const float* __restrict__ W,      // (96, K)
                         const float* __restrict__ bias,   // (96)
                         float* __restrict__ Z, int ldz, int zoff, int K) {
    int wave = threadIdx.x >> 5;
    int lane = threadIdx.x & 31;
    int tile = blockIdx.x * 4 + wave;
    int rowBase = tile * 16;
    int m  = lane & 15;
    int hi = lane >> 4;
    const float* Ap = A + (size_t)(rowBase + m) * lda + 4 * hi;
    const float* Wp = W + (size_t)m * K + 4 * hi;
    v8f acc[6] = {};

    v4f a0 = *(const v4f*)(Ap);
    v4f b0[6];
#pragma unroll
    for (int j = 0; j < 6; ++j) b0[j] = *(const v4f*)(Wp + (size_t)j * 16 * K);

    for (int k0 = 8; k0 < K; k0 += 8) {
        v4f a1 = *(const v4f*)(Ap + k0);
        v4f b1[6];
#pragma unroll
        for (int j = 0; j < 6; ++j)
            b1[j] = *(const v4f*)(Wp + (size_t)j * 16 * K + k0);
#pragma unroll
        for (int j = 0; j < 6; ++j) acc[j] = wmma_pair(acc[j], a0, b0[j]);
        a0 = a1;
#pragma unroll
        for (int j = 0; j < 6; ++j) b0[j] = b1[j];
    }
#pragma unroll
    for (int j = 0; j < 6; ++j) acc[j] = wmma_pair(acc[j], a0, b0[j]);

#pragma unroll
    for (int j = 0; j < 6; ++j) {
        int col = j * 16 + m;
        float bv = bias[col];
#pragma unroll
        for (int i = 0; i < 8; ++i) {
            int r = rowBase + i + 8 * hi;
            Z[(size_t)r * ldz + zoff + col] = acc[j][i] + bv;
        }
    }
}

// ---------------------------------------------------------------------------
// K9: fusion GEMM (K=192) with fused RevIN-denorm + (B,C,P)->(B,P,C) scatter.
// ---------------------------------------------------------------------------
__global__ void k_fus_out(const float* __restrict__ A,      // z (BC,192)
                          const float* __restrict__ W,      // fus_w (96,192)
                          const float* __restrict__ bias,   // fus_b
                          const float* __restrict__ rev_w,
                          const float* __restrict__ rev_b,
                          const float* __restrict__ meanv,
                          const float* __restrict__ stdv,
                          float* __restrict__ out) {
    const int K = 192;
    int wave = threadIdx.x >> 5;
    int lane = threadIdx.x & 31;
    int tile = blockIdx.x * 4 + wave;
    int rowBase = tile * 16;
    int m  = lane & 15;
    int hi = lane >> 4;
    const float* Ap = A + (size_t)(rowBase + m) * K + 4 * hi;
    const float* Wp = W + (size_t)m * K + 4 * hi;
    v8f acc[6] = {};

    v4f a0 = *(const v4f*)(Ap);
    v4f b0[6];
#pragma unroll
    for (int j = 0; j < 6; ++j) b0[j] = *(const v4f*)(Wp + (size_t)j * 16 * K);

    for (int k0 = 8; k0 < K; k0 += 8) {
        v4f a1 = *(const v4f*)(Ap + k0);
        v4f b1[6];
#pragma unroll
        for (int j = 0; j < 6; ++j)
            b1[j] = *(const v4f*)(Wp + (size_t)j * 16 * K + k0);
#pragma unroll
        for (int j = 0; j < 6; ++j) acc[j] = wmma_pair(acc[j], a0, b0[j]);
        a0 = a1;
#pragma unroll
        for (int j = 0; j < 6; ++j) b0[j] = b1[j];
    }
#pragma unroll
    for (int j = 0; j < 6; ++j) acc[j] = wmma_pair(acc[j], a0, b0[j]);

#pragma unroll
    for (int j = 0; j < 6; ++j) {
        int p = j * 16 + m;
        float bv = bias[p];
#pragma unroll
        for (int i = 0; i < 8; ++i) {
            int r = rowBase + i + 8 * hi;
            int b = r / Ck;
            int c = r - b * Ck;
            float val = acc[j][i] + bv;
            val = (val - rev_b[c]) / rev_w[c];
            out[((size_t)b * PREDk + p) * Ck + c] = val * stdv[r] + meanv[r];
        }
    }
}

// ---------------------------------------------------------------------------
extern "C" void kernel_launch(void* const* d_in, const int* in_sizes, int n_in,
                              void* d_out, int out_size, void* d_ws, size_t ws_size,
                              hipStream_t stream) {
    const float* x      = (const float*)d_in[0];
    const float* rev_w  = (const float*)d_in[1];
    const float* rev_b  = (const float*)d_in[2];
    const float* fc1_w  = (const float*)d_in[3];
    const float* fc1_b  = (const float*)d_in[4];
    const float* bn1_w  = (const float*)d_in[5];
    const float* bn1_b  = (const float*)d_in[6];
    const float* conv_w = (const float*)d_in[7];
    const float* conv_b = (const float*)d_in[8];
    const float* bn2_w  = (const float*)d_in[9];
    const float* bn2_b  = (const float*)d_in[10];
    const float* fc2_w  = (const float*)d_in[11];
    const float* fc2_b  = (const float*)d_in[12];
    const float* m1_w   = (const float*)d_in[13];
    const float* m1_b   = (const float*)d_in[14];
    const float* m2_w   = (const float*)d_in[15];
    const float* m2_b   = (const float*)d_in[16];
    const float* gl_sc  = (const float*)d_in[17];
    const float* ln_w   = (const float*)d_in[18];
    const float* ln_b   = (const float*)d_in[19];
    const float* seas_w = (const float*)d_in[20];
    const float* seas_b = (const float*)d_in[21];
    const float* tr_w   = (const float*)d_in[22];
    const float* tr_b   = (const float*)d_in[23];
    const float* fus_w  = (const float*)d_in[24];
    const float* fus_b  = (const float*)d_in[25];
    float* out = (float*)d_out;

    char* wp = (char*)d_ws;
    auto alloc = [&](size_t bytes) -> float* {
        float* p = (float*)wp;
        wp += (bytes + 255) & ~(size_t)255;
        return p;
    };
    float* s     = alloc((size_t)BCk * 512 * 4);
    float* trend = alloc((size_t)BCk * 512 * 4);
    float* meanv = alloc((size_t)BCk * 4);
    float* stdv  = alloc((size_t)BCk * 4);
    float* h     = alloc((size_t)BCk * 2048 * 4);
    float* pS    = alloc((size_t)64 * BCk * 4);
    float* pQ    = alloc((size_t)64 * BCk * 4);
    float* bn1S  = alloc(64 * 4);
    float* bn1O  = alloc(64 * 4);
    float* bn2S  = alloc(64 * 4);
    float* bn2O  = alloc(64 * 4);
    float* z     = alloc((size_t)BCk * 192 * 4);

    k_revin_ema<<<176, 128, 0, stream>>>(x, rev_w, rev_b, s, trend, meanv, stdv);
    k_fc1<<<BCk, 256, 0, stream>>>(s, fc1_w, fc1_b, h, pS, pQ);
    k_bnstats<<<64, 256, 0, stream>>>(pS, pQ, bn1_w, bn1_b, bn1S, bn1O);
    k_conv<<<(BCk * 64) / 256, 256, 0, stream>>>(h, bn1S, bn1O, conv_w, conv_b, pS, pQ);
    k_bnstats<<<64, 256, 0, stream>>>(pS, pQ, bn2_w, bn2_b, bn2S, bn2O);
    k_mix<<<BCk, 256, 0, stream>>>(s, h, bn2S, bn2O, fc2_w, fc2_b,
                                   m1_w, m1_b, m2_w, m2_b, gl_sc, ln_w, ln_b);
    // s_pred: y (stored in h rows, stride 2048) @ seas_w^T -> z[:,0:96]
    k_gemm96<<<321, 128, 0, stream>>>(h, 2048, seas_w, seas_b, z, 192, 0, 1024);
    // t_pred: trend @ tr_w^T -> z[:,96:192]
    k_gemm96<<<321, 128, 0, stream>>>(trend, 512, tr_w, tr_b, z, 192, 96, 512);
    // fusion GEMM + denorm + transpose
    k_fus_out<<<321, 128, 0, stream>>>(z, fus_w, fus_b, rev_w, rev_b, meanv, stdv, out);
}